// OpponentModelOracle_71116068487554
// MI455X (gfx1250) — compile-verified
//
#include <hip/hip_runtime.h>
#include <climits>

typedef __attribute__((ext_vector_type(2))) float v2f;
typedef __attribute__((ext_vector_type(4))) float v4f;
typedef __attribute__((ext_vector_type(8))) float v8f;

#define GRID_HW   4096   // 64*64 cells per batch element
#define GRID_W    64
#define NTHREADS  256
#define CELLS_PER_THREAD (GRID_HW / NTHREADS)   // 16

__global__ __launch_bounds__(NTHREADS)
void OpponentModelOracle_kernel(const float* __restrict__ x,
                                const float* __restrict__ projector,
                                float* __restrict__ out,
                                int nBatch)
{
    __shared__ unsigned int foodMask[GRID_HW / 32];   // 128 words = 512 B
    __shared__ int sFood;
    __shared__ int sOpp;
    __shared__ int sKey;

    const int b    = blockIdx.x;
    const int t    = threadIdx.x;
    const int lane = t & 31;
    const int wave = t >> 5;

    if (t == 0) { sFood = 0; sOpp = INT_MAX; sKey = INT_MAX; }
    __syncthreads();

    // ---- Pass 1: stream 4096 cells (16 B each) coalesced, build food bitmask,
    //              count food, find first opponent index (min row-major index).
    const v4f* cells = reinterpret_cast<const v4f*>(x) + (size_t)b * GRID_HW;

    int foodCount = 0;
    int minOpp    = INT_MAX;
    #pragma unroll
    for (int i = 0; i < CELLS_PER_THREAD; ++i) {
        const int cell = i * NTHREADS + t;
        v4f v = __builtin_nontemporal_load(&cells[cell]);
        const bool food = (v.y == 1.0f);   // channel 1
        const bool opp  = (v.w == 1.0f);   // channel 3
        // wave32 ballot: bit l of word <-> cell i*256 + wave*32 + l
        unsigned int bal = (unsigned int)__ballot(food);
        if (lane == 0) foodMask[i * 8 + wave] = bal;
        foodCount += food ? 1 : 0;
        if (opp) minOpp = min(minOpp, cell);
    }
    // wave32 reduction, then 8 LDS atomics
    #pragma unroll
    for (int o = 16; o; o >>= 1) {
        foodCount += __shfl_down(foodCount, o, 32);
        minOpp     = min(minOpp, __shfl_down(minOpp, o, 32));
    }
    if (lane == 0) { atomicAdd(&sFood, foodCount); atomicMin(&sOpp, minOpp); }
    __syncthreads();

    // ---- Opponent position (JAX argmax of all-False -> 0)
    const int oppIdx = (sOpp == INT_MAX) ? 0 : sOpp;
    const int ro = oppIdx >> 6;
    const int co = oppIdx & (GRID_W - 1);

    // ---- Pass 2: masked argmin of squared distance over food cells.
    //      key = (d2 << 12) | idx  gives min-d2 with first-index tie-break,
    //      exactly matching argmin over correctly-rounded sqrt distances.
    int key = INT_MAX;
    if (t < GRID_HW / 32) {
        unsigned int word = foodMask[t];
        while (word) {
            const int bit = __builtin_ctz(word);
            word &= word - 1;
            const int idx = t * 32 + bit;
            const int dr  = (idx >> 6) - ro;
            const int dc  = (idx & (GRID_W - 1)) - co;
            const int d2  = dr * dr + dc * dc;          // <= 7938, fits 13 bits
            key = min(key, (d2 << 12) | idx);
        }
    }
    #pragma unroll
    for (int o = 16; o; o >>= 1) key = min(key, __shfl_down(key, o, 32));
    if (lane == 0) atomicMin(&sKey, key);
    __syncthreads();

    // ---- Branch logic (uniform across block)
    const int  numFood    = sFood;
    const bool oppAtStart = (ro == 3) && (co == 6);
    const int  targetIdx  = (sKey == INT_MAX) ? 0 : (sKey & (GRID_HW - 1));
    const float isTop     = ((targetIdx >> 6) < 32) ? 1.0f : -1.0f;
    const float s = ((numFood > 1) && !oppAtStart) ? isTop
                  : ((numFood == 1) ? isTop : 0.0f);

    // ---- Tail: g[b, :] = s * projector[:] via V_WMMA_F32_16X16X4_F32 (wave 0).
    // A: s in column K=0 (VGPR0, lanes 0-15 = M 0..15), all other K exactly 0.
    // B: projector chunk in row K=0 (VGPR0, lanes 0-15 = N 0..15).
    // => D[m,n] = s * proj[n] exactly (zero terms contribute no rounding).
    if (t < 32) {
        v2f a;
        a.x = (lane < 16) ? s : 0.0f;   // K=0 (lanes 0-15) / K=2 (lanes 16-31)
        a.y = 0.0f;                     // K=1 / K=3
        #pragma unroll
        for (int j = 0; j < 8; ++j) {
            v2f bb;
            bb.x = (lane < 16) ? projector[j * 16 + lane] : 0.0f;
            bb.y = 0.0f;
            v8f c = {};
            c = __builtin_amdgcn_wmma_f32_16x16x4_f32(
                    /*neg_a=*/false, a, /*neg_b=*/false, bb,
                    /*c_mod=*/(short)0, c, /*reuse_a=*/false, /*reuse_b=*/false);
            if (lane < 16)   // VGPR0, lanes 0-15: D[M=0, N=lane]
                out[(size_t)b * 128 + j * 16 + lane] = c[0];
        }
    }

    // ---- Second output (zeros_like(g))
    if (t < 128)
        out[(size_t)nBatch * 128 + (size_t)b * 128 + t] = 0.0f;
}

extern "C" void kernel_launch(void* const* d_in, const int* in_sizes, int n_in,
                              void* d_out, int out_size, void* d_ws, size_t ws_size,
                              hipStream_t stream) {
    const float* x    = (const float*)d_in[0];      // (B, 64, 64, 4) f32
    const float* proj = (const float*)d_in[1];      // (128,) f32
    float* out        = (float*)d_out;              // (2, B, 128) f32 flat
    const int nBatch  = in_sizes[0] / (64 * 64 * 4);
    OpponentModelOracle_kernel<<<nBatch, NTHREADS, 0, stream>>>(x, proj, out, nBatch);
}